// MotionPrediction_69166153335040
// MI455X (gfx1250) — compile-verified
//
#include <hip/hip_runtime.h>
#include <hip/hip_bf16.h>

// ---------------------------------------------------------------------------
// Problem constants (from the reference setup)
// ---------------------------------------------------------------------------
#define BB 8
#define NN 4096
#define MM 4096
#define TT 10
#define NUM_HYPO 4
#define D2_THRESH 4.0f   // DIST_THRESH^2 (dist<2  <=>  max(d2,0)<4)
#define NSEG 4           // m-range split across the 4 waves of a block
#define SEG (MM / NSEG)  // 1024 boxes per wave

typedef __attribute__((ext_vector_type(2))) float v2f;
typedef __attribute__((ext_vector_type(8))) float v8f;

// streaming top-3 (ascending), strict <  (earlier m wins ties since m ascends)
__device__ __forceinline__ void insert3(float v, int m,
                                        float &t0, float &t1, float &t2,
                                        int &i0, int &i1, int &i2) {
  if (v < t2) {
    if (v < t1) {
      t2 = t1; i2 = i1;
      if (v < t0) { t1 = t0; i1 = i0; t0 = v; i0 = m; }
      else        { t1 = v;  i1 = m; }
    } else { t2 = v; i2 = m; }
  }
}

// lexicographic (d, idx) insertion for cross-partial merges (index tie-break,
// matching jax.lax.top_k's lowest-index-first rule on equal distances)
__device__ __forceinline__ void insert3x(float v, int m,
                                         float &t0, float &t1, float &t2,
                                         int &i0, int &i1, int &i2) {
  if ((v < t2) || (v == t2 && m < i2)) {
    if ((v < t1) || (v == t1 && m < i1)) {
      t2 = t1; i2 = i1;
      if ((v < t0) || (v == t0 && m < i0)) { t1 = t0; i1 = i0; t0 = v; i0 = m; }
      else                                 { t1 = v;  i1 = m; }
    } else { t2 = v; i2 = m; }
  }
}

// ---------------------------------------------------------------------------
// Kernel 1: fused pairwise-d2 (rank-4 GEMM via V_WMMA_F32_16X16X4_F32) +
// thresholded streaming top-3.
//   A (16x4, rows = boxes)      : [rx, ry, 1, |r|^2]
//   B (4x16, cols = detections) : [-2qx, -2qy, |q|^2, 1]
//   D[m,n] = |q|^2 + |r|^2 - 2 q.r = d2
// Each block owns one 16-detection tile; wave w scans m in [w*1024,(w+1)*1024)
// with a software-pipelined coordinate prefetch. Lists are initialized at
// D2_THRESH: candidates with d2 >= 4 can never affect the outputs (mask=0,
// id=M), so almost every tile is rejected by one min-tree compare.
// Partial top-3s merge lane<->lane+16 via shfl, then wave<->wave via LDS.
// ---------------------------------------------------------------------------
__global__ __launch_bounds__(128) void topk_dist_kernel(
    const float* __restrict__ tdet,    // (B,N,8)
    const float* __restrict__ boxes,   // (B,M,7)
    float* __restrict__ mask_out,      // (B,N,3) floats 1.0/0.0
    int*   __restrict__ matched)       // (B,N,3) ints (M = background)
{
  __shared__ float sD[NSEG][16][3];
  __shared__ int   sI[NSEG][16][3];

  const int lane = threadIdx.x & 31;
  const int wave = threadIdx.x >> 5;   // m-segment id
  const int b    = blockIdx.y;
  const int n0   = blockIdx.x * 16;
  const int col  = lane & 15;          // matrix column = detection within tile
  const int half = lane >> 4;          // 0: K=0,1 / rows 0-7 ; 1: K=2,3 / rows 8-15
  const int n    = n0 + col;

  // B operand (constant across the m-loop)
  const float qx = tdet[(size_t)(b * NN + n) * 8 + 0];
  const float qy = tdet[(size_t)(b * NN + n) * 8 + 1];
  v2f bop;
  if (half == 0) { bop.x = -2.0f * qx; bop.y = -2.0f * qy; }
  else           { bop.x = qx * qx + qy * qy; bop.y = 1.0f; }

  float t0 = D2_THRESH, t1 = D2_THRESH, t2 = D2_THRESH;
  int   i0 = MM,        i1 = MM,        i2 = MM;

  const int    wbase = wave * SEG;
  const float* bbox  = boxes + (size_t)b * MM * 7;

  // software pipeline: coords for tile `it` loaded during tile `it-1`
  float rx = bbox[(size_t)(wbase + col) * 7 + 0];
  float ry = bbox[(size_t)(wbase + col) * 7 + 1];

  for (int it = 0; it < SEG / 16; ++it) {
    int mnext = wbase + (it + 1) * 16 + col;
    mnext = mnext > MM - 1 ? MM - 1 : mnext;          // clamp: in-bounds prefetch
    const float nrx = bbox[(size_t)mnext * 7 + 0];
    const float nry = bbox[(size_t)mnext * 7 + 1];

    v2f aop;
    if (half == 0) { aop.x = rx;   aop.y = ry; }
    else           { aop.x = 1.0f; aop.y = rx * rx + ry * ry; }

    v8f c = {0.f, 0.f, 0.f, 0.f, 0.f, 0.f, 0.f, 0.f};
    v8f d = __builtin_amdgcn_wmma_f32_16x16x4_f32(
        /*neg_a=*/false, aop, /*neg_b=*/false, bop,
        /*c_mod=*/(short)0, c, /*reuse_a=*/false, /*reuse_b=*/false);

    // fast reject: only dive into insertion if some value beats the 3rd best
    const float mn = fminf(fminf(fminf(d[0], d[1]), fminf(d[2], d[3])),
                           fminf(fminf(d[4], d[5]), fminf(d[6], d[7])));
    if (mn < t2) {
      const int mbase = wbase + it * 16 + half * 8;   // VGPR i -> row i (+8 for half 1)
#pragma unroll
      for (int i = 0; i < 8; ++i)
        insert3(d[i], mbase + i, t0, t1, t2, i0, i1, i2);
    }
    rx = nrx; ry = nry;
  }

  // merge the two halves (lane L <-> L+16 hold disjoint m subsets of column n)
  float pt0 = __shfl_xor(t0, 16, 32); int pi0 = __shfl_xor(i0, 16, 32);
  float pt1 = __shfl_xor(t1, 16, 32); int pi1 = __shfl_xor(i1, 16, 32);
  float pt2 = __shfl_xor(t2, 16, 32); int pi2 = __shfl_xor(i2, 16, 32);
  insert3x(pt0, pi0, t0, t1, t2, i0, i1, i2);
  insert3x(pt1, pi1, t0, t1, t2, i0, i1, i2);
  insert3x(pt2, pi2, t0, t1, t2, i0, i1, i2);

  if (half == 0) {
    sD[wave][col][0] = t0; sD[wave][col][1] = t1; sD[wave][col][2] = t2;
    sI[wave][col][0] = i0; sI[wave][col][1] = i1; sI[wave][col][2] = i2;
  }
  __syncthreads();

  // final merge of the 4 per-wave partials (disjoint m-segments)
  if (threadIdx.x < 16) {
    const int cc = threadIdx.x;
    float u0 = D2_THRESH, u1 = D2_THRESH, u2 = D2_THRESH;
    int   j0 = MM,        j1 = MM,        j2 = MM;
#pragma unroll
    for (int w = 0; w < NSEG; ++w)
#pragma unroll
      for (int h = 0; h < 3; ++h)
        insert3x(sD[w][cc][h], sI[w][cc][h], u0, u1, u2, j0, j1, j2);

    const size_t base  = (size_t)(b * NN + n0 + cc) * 3;
    const float  dv[3] = {u0, u1, u2};
    const int    iv[3] = {j0, j1, j2};
#pragma unroll
    for (int h = 0; h < 3; ++h) {
      const bool valid = dv[h] < D2_THRESH;   // empty slots hold exactly 4.0
      mask_out[base + h] = valid ? 1.0f : 0.0f;
      matched[base + h]  = valid ? iv[h] : MM;
    }
  }
}

// ---------------------------------------------------------------------------
// Kernel 2: pure-bandwidth output assembly with float4 stores.
// One thread per (b, t, n): writes 32 contiguous floats of
// trajectory_hypotheses; t==0 additionally builds global_candidates
// (gather matched boxes, zero background row, time slot = 0).
// ---------------------------------------------------------------------------
__global__ __launch_bounds__(256) void assemble_kernel(
    const float* __restrict__ tdet,    // (B,N,8)
    const float* __restrict__ boxes,   // (B,M,7)
    const float* __restrict__ traj,    // (B,T,N,8)
    const int*   __restrict__ matched, // (B,N,3)
    float* __restrict__ trajh,         // (B,T+1,N,4,8)
    float* __restrict__ gcand)         // (B,1,N,4,8)
{
  const int idx   = blockIdx.x * blockDim.x + threadIdx.x;
  const int total = BB * (TT + 1) * NN;
  if (idx >= total) return;
  const int n  = idx % NN;
  const int bt = idx / NN;
  const int t  = bt % (TT + 1);
  const int b  = bt / (TT + 1);

  float4* out4 = (float4*)(trajh + (size_t)((b * (TT + 1) + t) * NN + n) * 32);

  if (t > 0) {
    const float4* src =
        (const float4*)(traj + (size_t)((b * TT + (t - 1)) * NN + n) * 8);
    const float4 v0 = src[0];
    const float4 v1 = src[1];
#pragma unroll
    for (int h = 0; h < NUM_HYPO; ++h) { out4[h * 2] = v0; out4[h * 2 + 1] = v1; }
  } else {
    float4* gc4 = (float4*)(gcand + (size_t)(b * NN + n) * 32);
    const float4* td4 = (const float4*)(tdet + (size_t)(b * NN + n) * 8);
    const float4 v0 = td4[0];
    const float4 v1 = td4[1];
    out4[0] = v0; out4[1] = v1;
    gc4[0]  = v0; gc4[1]  = v1;

    const int* mrow = matched + (size_t)(b * NN + n) * 3;
#pragma unroll
    for (int h = 1; h < NUM_HYPO; ++h) {
      const int id = mrow[h - 1];
      float vals[8] = {0.f, 0.f, 0.f, 0.f, 0.f, 0.f, 0.f, 0.f};
      if (id < MM) {
        const float* bp = boxes + (size_t)(b * MM + id) * 7;
#pragma unroll
        for (int c = 0; c < 7; ++c) vals[c] = bp[c];
        // vals[7] stays 0 (time slot)
      }
      const float4 w0 = make_float4(vals[0], vals[1], vals[2], vals[3]);
      const float4 w1 = make_float4(vals[4], vals[5], vals[6], vals[7]);
      out4[h * 2] = w0; out4[h * 2 + 1] = w1;
      gc4[h * 2]  = w0; gc4[h * 2 + 1]  = w1;
    }
  }
}

// ---------------------------------------------------------------------------
// Launch: d_in = {transfered_det, det_boxes3d, traj, num_hypo(scalar)}
// d_out = trajectory_hypotheses | global_candidates | valid_dist_mask (flat)
// d_ws  = matched ids, B*N*3 ints (384 KB)
// ---------------------------------------------------------------------------
extern "C" void kernel_launch(void* const* d_in, const int* in_sizes, int n_in,
                              void* d_out, int out_size, void* d_ws, size_t ws_size,
                              hipStream_t stream) {
  const float* tdet  = (const float*)d_in[0];
  const float* boxes = (const float*)d_in[1];
  const float* traj  = (const float*)d_in[2];
  (void)in_sizes; (void)n_in; (void)out_size; (void)ws_size;

  const size_t S1 = (size_t)BB * (TT + 1) * NN * NUM_HYPO * 8; // trajectory_hypotheses
  const size_t S2 = (size_t)BB * NN * NUM_HYPO * 8;            // global_candidates

  float* trajh   = (float*)d_out;
  float* gcand   = trajh + S1;
  float* maskout = trajh + S1 + S2;
  int*   matched = (int*)d_ws;

  // Kernel 1: one block per 16-detection tile; 4 waves split the m-range.
  dim3 g1(NN / 16, BB);
  topk_dist_kernel<<<g1, 128, 0, stream>>>(tdet, boxes, maskout, matched);

  // Kernel 2: one thread per (b, t, n), 32 floats each.
  const int total = BB * (TT + 1) * NN;
  assemble_kernel<<<(total + 255) / 256, 256, 0, stream>>>(
      tdet, boxes, traj, matched, trajh, gcand);
}